// ConvCapsMatrix_37434934952558
// MI455X (gfx1250) — compile-verified
//
#include <hip/hip_runtime.h>

// ============================================================================
// EM-routing capsule conv for MI455X (gfx1250).
//  Kernel 1: votes V via V_WMMA_F32_16X16X4_F32 (fp32 data -> f32 WMMA, K=4=y)
//            16 A-rows = 4 positions x 4 x-rows, 16 B-cols = 4 o x 4 z.
//  Kernel 2: 3 EM iterations per position, R/stats in LDS, V streamed from a
//            37.7MB chunk workspace that stays resident in the 192MB L2.
// ============================================================================

typedef __attribute__((ext_vector_type(2))) float v2f;
typedef __attribute__((ext_vector_type(8))) float v8f;

#define KHm 3
#define KWm 3
#define ICn 32
#define OCn 32
#define Pp  16
#define Nn  288          // KH*KW*IC
#define Hin 14
#define Win 14
#define HOo 12
#define WOo 12
#define NPOS 576         // 4*12*12
#define CHUNK 64         // positions per chunk (V chunk = 64*288*512*4B = 37.7MB, L2-resident)
#define NCHUNK 9
#define GPC 16           // groups-of-4-positions per chunk (CHUNK/4)
#define EPSf 1e-8f

// ---------------------------------------------------------------------------
// Kernel 1: V[(posLocal*288+n)*512 + o*16 + x*4 + z] = sum_y xp * W
// one wave per (n, group-of-4-positions); 8 WMMAs sweep o in chunks of 4.
// ---------------------------------------------------------------------------
__global__ __launch_bounds__(256)
void caps_v_wmma(const float* __restrict__ xin, const float* __restrict__ Wt,
                 float* __restrict__ Vws, int chunk)
{
    const int tid  = threadIdx.x;
    const int lane = tid & 31;
    const int wid  = blockIdx.x * 8 + (tid >> 5);   // 0 .. GPC*Nn-1
    const int gl   = wid % GPC;                     // local group in chunk
    const int n    = wid / GPC;                     // 0..287
    const int g    = chunk * GPC + gl;              // global group 0..143
    const int k = n / 96, l = (n / 32) % 3, c = n & 31;
    const int b = g / 36;
    const int h = (g % 36) / 3;
    const int wbase = (g % 3) * 4;                  // 4 consecutive w positions

    // ---- A fragment (16x4 f32): row M = lane&15 = pig*4 + x ; K half = lane>>4
    const int Ma  = lane & 15;
    const int kk  = lane >> 4;                      // K base = 2*kk
    const int pig = Ma >> 2;                        // position in group
    const int xr  = Ma & 3;                         // x row of 4x4 pose
    const int w   = wbase + pig;
    const float* apx = xin + ((((size_t)b * Hin + (h + k)) * Win + (w + l)) * ICn + c) * Pp
                           + xr * 4 + kk * 2;
    v2f afrag;
    afrag.x = apx[0];                               // y = 2*kk
    afrag.y = apx[1];                               // y = 2*kk+1

    // ---- B fragment (4x16 f32): col N = lane&15 = oLocal*4 + z
    const int Nb  = lane & 15;
    const int zb  = Nb & 3;
    const int obl = Nb >> 2;
    const float* wrow = Wt + ((((size_t)k * KWm + l) * ICn + c) * OCn) * Pp;

#pragma unroll
    for (int t = 0; t < 8; ++t) {
        const int o = t * 4 + obl;
        const float* wp = wrow + (size_t)o * Pp + (kk * 2) * 4 + zb;
        v2f bfrag;
        bfrag.x = wp[0];                            // y = 2*kk
        bfrag.y = wp[4];                            // y = 2*kk+1
        v8f acc = {};
        acc = __builtin_amdgcn_wmma_f32_16x16x4_f32(
            /*neg_a=*/false, afrag, /*neg_b=*/false, bfrag,
            /*c_mod=*/(short)0, acc, /*reuse_a=*/false, /*reuse_b=*/false);
        // D layout: VGPR r, lanes 0-15: M=r ; lanes 16-31: M=r+8 ; N = lane&15
#pragma unroll
        for (int r = 0; r < 8; ++r) {
            const int Md   = r + (lane >> 4) * 8;
            const int pigd = Md >> 2, xd = Md & 3;
            const int posL = gl * 4 + pigd;
            const size_t idx = (((size_t)posL * Nn + n) * 512)
                             + (size_t)o * Pp + xd * 4 + zb;
            Vws[idx] = acc[r];
        }
    }
}

// ---------------------------------------------------------------------------
// Kernel 2: 3 EM iterations per output position. One block per position.
// ---------------------------------------------------------------------------
#define RSTRIDE 33       // pad R rows -> conflict-free per-o writes

__global__ __launch_bounds__(256)
void caps_route(const float* __restrict__ ain, const float* __restrict__ beta_u,
                const float* __restrict__ beta_a, const float* __restrict__ Vws,
                float* __restrict__ out, int chunk)
{
    __shared__ float sR[Nn * RSTRIDE];   // routing weights R[n][o]
    __shared__ float sAin[Nn];
    __shared__ float sMu[512];
    __shared__ float sSig[512];
    __shared__ float sInv[512];
    __shared__ float sRs[OCn];
    __shared__ float sLsum[OCn];
    __shared__ float sAout[OCn];

    const int tid = threadIdx.x;
    const int pos = chunk * CHUNK + blockIdx.x;
    const int b = pos / (HOo * WOo);
    const int h = (pos / WOo) % HOo;
    const int w = pos % WOo;
    const float* __restrict__ Vp = Vws + (size_t)blockIdx.x * Nn * 512;

    for (int i = tid; i < Nn * RSTRIDE; i += 256) sR[i] = 1.0f / OCn;
    for (int i = tid; i < Nn; i += 256) {
        const int k = i / 96, l = (i / 32) % 3, c = i & 31;
        sAin[i] = ain[(((size_t)b * Hin + (h + k)) * Win + (w + l)) * ICn + c];
    }
    __syncthreads();

    const int op0 = tid, op1 = tid + 256;           // each thread owns 2 (o,p)
    const int o0 = op0 >> 4, o1 = op1 >> 4;
    float mu0 = 0.0f, mu1 = 0.0f;

    for (int it = 0; it < 3; ++it) {
        // ---- pass A: weighted 1st/2nd moments over n (coalesced V reads)
        float s1a = 0, s2a = 0, s1b = 0, s2b = 0;
        for (int n = 0; n < Nn; ++n) {
            const float an  = sAin[n];
            const float ra0 = sR[n * RSTRIDE + o0] * an;
            const float ra1 = sR[n * RSTRIDE + o1] * an;
            const float v0  = Vp[(size_t)n * 512 + op0];
            const float v1  = Vp[(size_t)n * 512 + op1];
            s1a = fmaf(ra0, v0, s1a);  s2a = fmaf(ra0 * v0, v0, s2a);
            s1b = fmaf(ra1, v1, s1b);  s2b = fmaf(ra1 * v1, v1, s2b);
        }
        if (tid < OCn) {
            float rs = 0.0f;
            for (int n = 0; n < Nn; ++n) rs += sR[n * RSTRIDE + tid] * sAin[n];
            sRs[tid] = rs + EPSf;
        }
        __syncthreads();

        const float irs0 = 1.0f / sRs[o0], irs1 = 1.0f / sRs[o1];
        mu0 = s1a * irs0;
        mu1 = s1b * irs1;
        const float sg0 = s2a * irs0 - mu0 * mu0 + EPSf;
        const float sg1 = s2b * irs1 - mu1 * mu1 + EPSf;
        sMu[op0] = mu0;  sSig[op0] = sg0;  sInv[op0] = 1.0f / sg0;
        sMu[op1] = mu1;  sSig[op1] = sg1;  sInv[op1] = 1.0f / sg1;
        __syncthreads();

        // ---- activation cost per o
        if (tid < OCn) {
            float ls = 0.0f;
            for (int p = 0; p < Pp; ++p) ls += __logf(sSig[tid * Pp + p]);
            sLsum[tid] = ls;
            const float cost = sRs[tid] * (16.0f * beta_u[tid] + 0.5f * ls);
            sAout[tid] = 1.0f / (1.0f + __expf(-(beta_a[tid] - cost)));
        }
        __syncthreads();
        if (it == 2) break;

        // ---- R update: each thread owns whole n rows, softmax over o
        for (int n = tid; n < Nn; n += 256) {
            float logit[OCn];
            float mx = -3.4e38f;
            const float* vrow = Vp + (size_t)n * 512;
#pragma unroll
            for (int o = 0; o < OCn; ++o) {
                float s = 0.0f;
#pragma unroll
                for (int p = 0; p < Pp; ++p) {
                    const float d = vrow[o * Pp + p] - sMu[o * Pp + p];
                    s = fmaf(d * d, sInv[o * Pp + p], s);
                }
                const float lg = __logf(sAout[o] + EPSf) - 0.5f * (s + sLsum[o]);
                logit[o] = lg;
                mx = fmaxf(mx, lg);
            }
            float se = 0.0f;
#pragma unroll
            for (int o = 0; o < OCn; ++o) {
                const float e = __expf(logit[o] - mx);
                logit[o] = e;
                se += e;
            }
            const float ise = 1.0f / se;
#pragma unroll
            for (int o = 0; o < OCn; ++o) sR[n * RSTRIDE + o] = logit[o] * ise;
        }
        __syncthreads();
    }

    // ---- outputs: pose (B,Ho,Wo,OC,4,4) then a_out (B,Ho,Wo,OC)
    out[(size_t)pos * 512 + op0] = mu0;
    out[(size_t)pos * 512 + op1] = mu1;
    if (tid < OCn) out[(size_t)NPOS * 512 + (size_t)pos * OCn + tid] = sAout[tid];
}

// ---------------------------------------------------------------------------
extern "C" void kernel_launch(void* const* d_in, const int* in_sizes, int n_in,
                              void* d_out, int out_size, void* d_ws, size_t ws_size,
                              hipStream_t stream)
{
    const float* x      = (const float*)d_in[0];   // (4,14,14,32,4,4)
    const float* a      = (const float*)d_in[1];   // (4,14,14,32)
    const float* W      = (const float*)d_in[2];   // (3,3,32,32,4,4)
    const float* beta_u = (const float*)d_in[3];   // (32,)
    const float* beta_a = (const float*)d_in[4];   // (32,)
    float* out = (float*)d_out;
    float* Vws = (float*)d_ws;   // needs CHUNK*288*512*4 = 37,748,736 bytes

    for (int ch = 0; ch < NCHUNK; ++ch) {
        caps_v_wmma<<<GPC * Nn / 8, 256, 0, stream>>>(x, W, Vws, ch);
        caps_route<<<CHUNK, 256, 0, stream>>>(a, beta_u, beta_a, Vws, out, ch);
    }
}